// TruePixelwiseMLPRelight_816043786368
// MI455X (gfx1250) — compile-verified
//
#include <hip/hip_runtime.h>

typedef float v2f __attribute__((ext_vector_type(2)));
typedef float v4f __attribute__((ext_vector_type(4)));
typedef float v8f __attribute__((ext_vector_type(8)));

#define NPATCH 16384
#define NBATCH 32

// ---------------------------------------------------------------------------
// Phase 1: per-patch MLP (2->8->8->8->3, relu/sigmoid) via V_WMMA_F32_16X16X4.
// One wave per patch; batch dim (32) = two 16-row M tiles; neurons = N (padded
// to 16); K padded 2->4 for layer 1 and split 8 = 2x4 for hidden layers.
// Layout conversion D-layout -> A-layout goes through per-wave LDS (DS ops are
// in-order within a wave, so no barriers are required).
// All operand loads are branch-free: unconditional float2 loads at clamped
// (always in-bounds) addresses + v_cndmask selects for the zero padding.
// ---------------------------------------------------------------------------
__global__ __launch_bounds__(256) void mlp_patch_kernel(
    const float* __restrict__ xdir,                         // [32][2]
    const float* __restrict__ w1, const float* __restrict__ b1,
    const float* __restrict__ w2, const float* __restrict__ b2,
    const float* __restrict__ w3, const float* __restrict__ b3,
    const float* __restrict__ wo, const float* __restrict__ bo,
    float* __restrict__ ws)                                 // [32][16384][4]
{
  __shared__ float hbuf[8][2][NBATCH * 16];  // 8 waves x ping-pong x [batch][16]

  const int tid  = threadIdx.x;
  const int lane = tid & 31;
  const int wv   = tid >> 5;
  const int p    = blockIdx.x * 8 + wv;      // patch id [0, 16384)

  const int half = lane >> 4;                // 0: lanes 0-15, 1: lanes 16-31
  const int n    = lane & 15;                // N column (or M row for A tiles)
  const int nc   = (n < 8) ? n : 0;          // clamped neuron index
  const int nc3  = (n < 3) ? n : 0;          // clamped output-channel index

  const float* W1 = w1 + p * 16;             // [8][2]
  const float* W2 = w2 + p * 64;             // [8][8]
  const float* W3 = w3 + p * 64;             // [8][8]
  const float* WO = wo + p * 24;             // [3][8]

  // branch-free bias loads (unconditional load at clamped index, then select)
  const float lb1 = b1[p * 8 + nc];
  const float lb2 = b2[p * 8 + nc];
  const float lb3 = b3[p * 8 + nc];
  const float lbo = bo[p * 3 + nc3];
  const float bias1 = (n < 8) ? lb1 : 0.0f;
  const float bias2 = (n < 8) ? lb2 : 0.0f;
  const float bias3 = (n < 8) ? lb3 : 0.0f;
  const float biaso = (n < 3) ? lbo : 0.0f;

  float* h0 = &hbuf[wv][0][0];
  float* h1 = &hbuf[wv][1][0];

  // ---- Layer 1: D[32x16] = X[32x2 pad 4] * W1^T[4x16 pad] ----
  // B operand: lanes<16 hold rows K=0,1 (real), lanes>=16 rows K=2,3 (zero).
  const v2f lw1 = *(const v2f*)(W1 + nc * 2);   // 8B aligned
  v2f bm1;
  bm1.x = (half == 0 && n < 8) ? lw1.x : 0.0f;
  bm1.y = (half == 0 && n < 8) ? lw1.y : 0.0f;

  #pragma unroll
  for (int t = 0; t < 2; ++t) {              // batch tiles 0-15 / 16-31
    const int m = n + 16 * t;                // always < 32: safe to load
    const v2f lx = *(const v2f*)(xdir + m * 2);
    v2f a;
    a.x = (half == 0) ? lx.x : 0.0f;         // K=0 (zero-pad K=2 on upper half)
    a.y = (half == 0) ? lx.y : 0.0f;         // K=1 (zero-pad K=3 on upper half)
    v8f d = {};
    d = __builtin_amdgcn_wmma_f32_16x16x4_f32(false, a, false, bm1,
                                              (short)0, d, false, false);
    #pragma unroll
    for (int v = 0; v < 8; ++v) {
      float val = fmaxf(d[v] + bias1, 0.0f);
      const int batch = v + 8 * half + 16 * t;
      h0[batch * 16 + n] = val;              // cols 8..15 store exact zeros
    }
  }

  // ---- Hidden layer (K=8): D[32x16] = H[32x8] * W^T, two K=4 chunks ----
  auto layer8 = [&](const float* hin, const float* W, float bias, float* hout) {
    #pragma unroll
    for (int t = 0; t < 2; ++t) {
      v8f d = {};
      #pragma unroll
      for (int c = 0; c < 2; ++c) {
        const int kb = 4 * c + 2 * half;     // K base handled by this lane half
        const int m  = n + 16 * t;
        const v2f lw = *(const v2f*)(W + nc * 8 + kb);  // 8B aligned
        v2f a, b;
        a.x = hin[m * 16 + kb + 0];
        a.y = hin[m * 16 + kb + 1];
        b.x = (n < 8) ? lw.x : 0.0f;
        b.y = (n < 8) ? lw.y : 0.0f;
        d = __builtin_amdgcn_wmma_f32_16x16x4_f32(false, a, false, b,
                                                  (short)0, d, false, false);
      }
      #pragma unroll
      for (int v = 0; v < 8; ++v) {
        float val = fmaxf(d[v] + bias, 0.0f);
        const int batch = v + 8 * half + 16 * t;
        hout[batch * 16 + n] = val;
      }
    }
  };

  layer8(h0, W2, bias2, h1);
  layer8(h1, W3, bias3, h0);

  // ---- Output layer (K=8, N=3) + sigmoid, write colors to workspace ----
  #pragma unroll
  for (int t = 0; t < 2; ++t) {
    v8f d = {};
    #pragma unroll
    for (int c = 0; c < 2; ++c) {
      const int kb = 4 * c + 2 * half;
      const int m  = n + 16 * t;
      const v2f lw = *(const v2f*)(WO + nc3 * 8 + kb);  // 8B aligned
      v2f a, b;
      a.x = h0[m * 16 + kb + 0];
      a.y = h0[m * 16 + kb + 1];
      b.x = (n < 3) ? lw.x : 0.0f;
      b.y = (n < 3) ? lw.y : 0.0f;
      d = __builtin_amdgcn_wmma_f32_16x16x4_f32(false, a, false, b,
                                                (short)0, d, false, false);
    }
    #pragma unroll
    for (int v = 0; v < 8; ++v) {
      if (n < 4) {                           // store rgb + pad channel
        const float z = d[v] + biaso;
        const float s = 1.0f / (1.0f + __expf(-z));
        const int batch = v + 8 * half + 16 * t;
        ws[(batch * NPATCH + p) * 4 + n] = s;
      }
    }
  }
}

// ---------------------------------------------------------------------------
// Phase 2: tile each patch color to 8x8 pixels. Pure 128-bit streaming stores:
// 402.6 MB total -> this kernel IS the runtime (HBM-store bound, ~17 us floor).
// One 16B vector per thread; each 256-thread block writes 4 KB contiguous.
// Nontemporal stores: output is write-once, keep it from rinsing L2.
// Row layout per patch-col: r g b r | g b r g | b r g b | (repeats).
// ---------------------------------------------------------------------------
__global__ __launch_bounds__(256) void tile_write_kernel(
    const v4f* __restrict__ ws4,             // [32][16384] (rgb pad)
    v4f* __restrict__ out4)                  // [32][1024][768]
{
  const int j  = blockIdx.x * 256 + threadIdx.x;  // vec4 index in row [0,768)
  const int r  = blockIdx.y;                      // image row
  const int b  = blockIdx.z;                      // batch
  const int ph = r >> 3;
  const int pw = j / 6;                           // 6 vec4 per patch column
  const int jj = j - pw * 6;

  const v4f c = ws4[b * NPATCH + ph * 128 + pw];
  v4f o;
  const int phase = jj % 3;                       // (jj*4) % 3 == jj % 3
  if (phase == 0)      o = (v4f){c.x, c.y, c.z, c.x};
  else if (phase == 1) o = (v4f){c.y, c.z, c.x, c.y};
  else                 o = (v4f){c.z, c.x, c.y, c.z};

  __builtin_nontemporal_store(o, &out4[(b * 1024 + r) * 768 + j]);
}

extern "C" void kernel_launch(void* const* d_in, const int* in_sizes, int n_in,
                              void* d_out, int out_size, void* d_ws, size_t ws_size,
                              hipStream_t stream) {
  const float* x  = (const float*)d_in[0];
  const float* w1 = (const float*)d_in[1];
  const float* b1 = (const float*)d_in[2];
  const float* w2 = (const float*)d_in[3];
  const float* b2 = (const float*)d_in[4];
  const float* w3 = (const float*)d_in[5];
  const float* b3 = (const float*)d_in[6];
  const float* wo = (const float*)d_in[7];
  const float* bo = (const float*)d_in[8];
  float* ws = (float*)d_ws;  // needs 32*16384*4 floats = 8 MB (L2-resident)

  mlp_patch_kernel<<<NPATCH / 8, 256, 0, stream>>>(x, w1, b1, w2, b2, w3, b3,
                                                   wo, bo, ws);
  tile_write_kernel<<<dim3(3, 1024, 32), 256, 0, stream>>>(
      (const v4f*)ws, (v4f*)d_out);
}